// REGINConv_23553600651700
// MI455X (gfx1250) — compile-verified
//
#include <hip/hip_runtime.h>

typedef float v2f __attribute__((ext_vector_type(2)));
typedef float v8f __attribute__((ext_vector_type(8)));

#define NN 100000     // nodes
#define NE 3200000    // edges
#define DD 64         // feature dim
#define ALPHA_F 10.0f
#define NEG_SLOPE_F 0.01f

// --- per-edge weight: LeakyReLU(edge_weight[e_feat-1] * alpha) -----------
__device__ __forceinline__ float leaky_ew(const float* __restrict__ ew_tab,
                                          const int* __restrict__ efeat, int e) {
  float x = ew_tab[efeat[e] - 1] * ALPHA_F;
  return x >= 0.0f ? x : x * NEG_SLOPE_F;
}

// --- pass 1: weighted in-degree per dst node -----------------------------
__global__ void deg_kernel(const float* __restrict__ ew_tab,
                           const int* __restrict__ efeat,
                           const int* __restrict__ dst,
                           float* __restrict__ deg) {
  int e = blockIdx.x * blockDim.x + threadIdx.x;
  if (e < NE) {
    unsafeAtomicAdd(&deg[dst[e]], leaky_ew(ew_tab, efeat, e));
  }
}

// --- pass 2: norm = rsqrt(max(deg,1)), in place --------------------------
__global__ void norm_kernel(float* __restrict__ deg) {
  int i = blockIdx.x * blockDim.x + threadIdx.x;
  if (i < NN) deg[i] = rsqrtf(fmaxf(deg[i], 1.0f));
}

// --- pass 3: one wave32 per edge; each lane moves a float2 of the row ----
// agg[dst] += feat[src] * (norm[src] * ew)   (h = feat*norm fused in)
__global__ void agg_kernel(const float* __restrict__ feat,
                           const float* __restrict__ ew_tab,
                           const int* __restrict__ efeat,
                           const int* __restrict__ src,
                           const int* __restrict__ dst,
                           const float* __restrict__ norm,
                           float* __restrict__ agg) {
  int wave = threadIdx.x >> 5;
  int lane = threadIdx.x & 31;
  int e = blockIdx.x * 8 + wave;      // blockDim = 256 -> 8 waves
  if (e >= NE) return;
  int s = src[e];
  int d = dst[e];
  float w = leaky_ew(ew_tab, efeat, e) * norm[s];
  float2 v = ((const float2*)(feat + (size_t)s * DD))[lane];
  float* a = agg + (size_t)d * DD + lane * 2;
  unsafeAtomicAdd(a,     v.x * w);
  unsafeAtomicAdd(a + 1, v.y * w);
}

// --- pass 4: rst = agg*norm ; hid = relu(rst@W1+b1) ; out = hid@W2+b2 ----
// 4 waves/block, each wave owns a 16-row tile. W1/W2/b1/b2 staged in LDS.
// WMMA: V_WMMA_F32_16X16X4_F32, K=64 -> 16 k-steps, 4 column tiles.
#define MLP_WAVES 4
#define HSTRIDE 65   // padded row stride for hid tile (bank-conflict free)

__launch_bounds__(128, 1)
__global__ void mlp_kernel(const float* __restrict__ agg,
                           const float* __restrict__ norm,
                           const float* __restrict__ W1,
                           const float* __restrict__ b1,
                           const float* __restrict__ W2,
                           const float* __restrict__ b2,
                           float* __restrict__ out) {
  __shared__ float sW1[DD * DD];
  __shared__ float sW2[DD * DD];
  __shared__ float sB1[DD];
  __shared__ float sB2[DD];
  __shared__ float sHid[MLP_WAVES][16 * HSTRIDE];

  int tid = threadIdx.x;
  for (int i = tid; i < DD * DD; i += 128) { sW1[i] = W1[i]; sW2[i] = W2[i]; }
  if (tid < DD) { sB1[tid] = b1[tid]; sB2[tid] = b2[tid]; }
  __syncthreads();

  int wave  = tid >> 5;
  int lane  = tid & 31;
  int r     = lane & 15;     // row within 16-row tile (A) / col within tile (B,C)
  int khalf = lane >> 4;     // 0: K={0,1}, 1: K={2,3} within each 4-wide k-step
  int rowBase = (blockIdx.x * MLP_WAVES + wave) * 16;
  int row   = rowBase + r;
  bool valid = row < NN;
  int  srow  = valid ? row : 0;
  float sc   = valid ? norm[srow] : 0.0f;   // zero contribution from pad rows

  // ---- A fragments for layer 1: rst row = agg[row] * norm[row] ----
  v2f afrag[16];
  const float* arow = agg + (size_t)srow * DD;
  #pragma unroll
  for (int kk = 0; kk < 16; ++kk) {
    int c = kk * 4 + khalf * 2;
    v2f a; a.x = arow[c] * sc; a.y = arow[c + 1] * sc;
    afrag[kk] = a;
  }

  // ---- layer 1: hid = relu(rst @ W1 + b1) ----
  v8f acc[4];
  #pragma unroll
  for (int nt = 0; nt < 4; ++nt) {
    v8f c = {};
    #pragma unroll
    for (int kk = 0; kk < 16; ++kk) {
      int kb = kk * 4 + khalf * 2;
      v2f b; b.x = sW1[kb * DD + nt * 16 + r]; b.y = sW1[(kb + 1) * DD + nt * 16 + r];
      c = __builtin_amdgcn_wmma_f32_16x16x4_f32(false, afrag[kk], false, b,
                                                (short)0, c, false, false);
    }
    acc[nt] = c;
  }

  // C-layout -> LDS tile (row-major, padded) with bias + ReLU fused
  float* hid = &sHid[wave][0];
  #pragma unroll
  for (int nt = 0; nt < 4; ++nt) {
    int col = nt * 16 + r;
    float bb = sB1[col];
    #pragma unroll
    for (int j = 0; j < 8; ++j) {
      int m = j + khalf * 8;
      float v = acc[nt][j] + bb;
      hid[m * HSTRIDE + col] = v > 0.0f ? v : 0.0f;
    }
  }
  __syncthreads();

  // ---- A fragments for layer 2 from LDS hid tile ----
  #pragma unroll
  for (int kk = 0; kk < 16; ++kk) {
    int c = kk * 4 + khalf * 2;
    v2f a; a.x = hid[r * HSTRIDE + c]; a.y = hid[r * HSTRIDE + c + 1];
    afrag[kk] = a;
  }

  // ---- layer 2: out = hid @ W2 + b2 ----
  #pragma unroll
  for (int nt = 0; nt < 4; ++nt) {
    v8f c = {};
    #pragma unroll
    for (int kk = 0; kk < 16; ++kk) {
      int kb = kk * 4 + khalf * 2;
      v2f b; b.x = sW2[kb * DD + nt * 16 + r]; b.y = sW2[(kb + 1) * DD + nt * 16 + r];
      c = __builtin_amdgcn_wmma_f32_16x16x4_f32(false, afrag[kk], false, b,
                                                (short)0, c, false, false);
    }
    acc[nt] = c;
  }

  // ---- store (per-lane guard only after all WMMA issued) ----
  #pragma unroll
  for (int nt = 0; nt < 4; ++nt) {
    int col = nt * 16 + r;
    float bb = sB2[col];
    #pragma unroll
    for (int j = 0; j < 8; ++j) {
      int m = j + khalf * 8;
      int orow = rowBase + m;
      if (orow < NN) out[(size_t)orow * DD + col] = acc[nt][j] + bb;
    }
  }
}

extern "C" void kernel_launch(void* const* d_in, const int* in_sizes, int n_in,
                              void* d_out, int out_size, void* d_ws, size_t ws_size,
                              hipStream_t stream) {
  const float* feat  = (const float*)d_in[0];   // [N, 64]
  const float* ewtab = (const float*)d_in[1];   // [8, 1]
  const float* W1    = (const float*)d_in[2];   // [64, 64]
  const float* b1    = (const float*)d_in[3];   // [64]
  const float* W2    = (const float*)d_in[4];   // [64, 64]
  const float* b2    = (const float*)d_in[5];   // [64]
  const int*   src   = (const int*)d_in[6];     // [E]
  const int*   dst   = (const int*)d_in[7];     // [E]
  const int*   efeat = (const int*)d_in[8];     // [E], values 1..8
  float* out = (float*)d_out;                   // [N, 64]

  float* deg = (float*)d_ws;                    // N floats (becomes norm)
  float* agg = deg + NN;                        // N*64 floats

  hipMemsetAsync(d_ws, 0, (size_t)(NN + (size_t)NN * DD) * sizeof(float), stream);

  deg_kernel<<<(NE + 255) / 256, 256, 0, stream>>>(ewtab, efeat, dst, deg);
  norm_kernel<<<(NN + 255) / 256, 256, 0, stream>>>(deg);
  agg_kernel<<<(NE + 7) / 8, 256, 0, stream>>>(feat, ewtab, efeat, src, dst, deg, agg);
  mlp_kernel<<<(NN + MLP_WAVES * 16 - 1) / (MLP_WAVES * 16), 128, 0, stream>>>(
      agg, deg, W1, b1, W2, b2, out);
}